// SAGENet_71588514890204
// MI455X (gfx1250) — compile-verified
//
#include <hip/hip_runtime.h>
#include <math.h>

typedef __attribute__((ext_vector_type(2))) float v2f;
typedef __attribute__((ext_vector_type(2))) unsigned int v2u;
typedef __attribute__((ext_vector_type(8))) float v8f;

__device__ __forceinline__ int imin(int a, int b) { return a < b ? a : b; }

// ---------------------------------------------------------------------------
// Order-preserving float <-> uint key transform (radix trick). Allows a single
// branchless atomicMax(u32) to implement float max. No NaNs in this workload.
//   key(-inf) = 0x007FFFFF  (buffer init value == "empty segment" marker)
// ---------------------------------------------------------------------------
#define KEY_NEG_INF 0x007FFFFFu

__device__ __forceinline__ unsigned int f32_to_key(float v) {
  unsigned int u = __float_as_uint(v);
  return (u & 0x80000000u) ? ~u : (u | 0x80000000u);
}
__device__ __forceinline__ float key_to_f32(unsigned int k) {
  unsigned int u = (k & 0x80000000u) ? (k & 0x7FFFFFFFu) : ~k;
  return __uint_as_float(u);
}

// ds_swizzle_b32 xor-swap within 32-lane groups (offset is an ICE via template)
template <int PAT>
__device__ __forceinline__ float swz(float v) {
  return __int_as_float(__builtin_amdgcn_ds_swizzle(__float_as_int(v), PAT));
}
#define SWZ_X1 0x041F
#define SWZ_X2 0x081F
#define SWZ_X4 0x101F
#define SWZ_X8 0x201F

// ---------------------------------------------------------------------------
// Fill buffer with a constant uint pattern (key(-inf) init)
// ---------------------------------------------------------------------------
__global__ void fill_kernel(unsigned int* __restrict__ p, unsigned int v, long n) {
  long i = (long)blockIdx.x * blockDim.x + threadIdx.x;
  long stride = (long)gridDim.x * blockDim.x;
  for (; i < n; i += stride) p[i] = v;
}

// ---------------------------------------------------------------------------
// In-place fixup: key buffer -> plain floats, with key(-inf) (empty segment)
// mapped to 0. One streaming pass so the GEMM loop needs zero per-element VALU
// and re-reads (once per column tile) stay clean.
// ---------------------------------------------------------------------------
__global__ void fixup_kernel(unsigned int* __restrict__ p, long n2 /* pairs */) {
  long i = (long)blockIdx.x * blockDim.x + threadIdx.x;
  long stride = (long)gridDim.x * blockDim.x;
  v2u* __restrict__ p2 = (v2u*)p;
  for (; i < n2; i += stride) {
    v2u k = p2[i];
    float f0 = key_to_f32(k.x);
    float f1 = key_to_f32(k.y);
    v2u o;
    o.x = __float_as_uint(f0 == -INFINITY ? 0.0f : f0);
    o.y = __float_as_uint(f1 == -INFINITY ? 0.0f : f1);
    p2[i] = o;
  }
}

// ---------------------------------------------------------------------------
// Transpose W [K, Nout] -> Wt [Nout, K] so B fragments load as aligned b64
// pairs along K. W matrices are tiny (<= 25600 elements).
// ---------------------------------------------------------------------------
__global__ void transpose_kernel(const float* __restrict__ w,
                                 float* __restrict__ wt, int K, int Nout) {
  int idx = blockIdx.x * blockDim.x + threadIdx.x;
  if (idx >= K * Nout) return;
  int k = idx / Nout;
  int n = idx % Nout;
  wt[(long)n * K + k] = w[idx];
}

// ---------------------------------------------------------------------------
// Scatter-max: one wave per edge; lanes stride the feature row in float2
// chunks (F is even, rows are 8B aligned). One u32 atomicMax per element.
// ---------------------------------------------------------------------------
__global__ __launch_bounds__(256) void scatter_max_kernel(
    const float* __restrict__ feat, const int* __restrict__ src,
    const int* __restrict__ dst, unsigned int* __restrict__ agg, int E, int F) {
  const int wavesPerBlock = blockDim.x >> 5;
  const int e = blockIdx.x * wavesPerBlock + (threadIdx.x >> 5);
  if (e >= E) return;  // wave-uniform
  const int lane = threadIdx.x & 31;
  const v2f* __restrict__ s2 = (const v2f*)(feat + (long)src[e] * F);
  unsigned int* __restrict__ d = agg + (long)dst[e] * F;
  const int chunks = F >> 1;
  for (int c = lane; c < chunks; c += 32) {
    v2f v = s2[c];
    atomicMax(&d[2 * c], f32_to_key(v.x));
    atomicMax(&d[2 * c + 1], f32_to_key(v.y));
  }
}

// ---------------------------------------------------------------------------
// Fused SAGE layer GEMM using fp32 WMMA (V_WMMA_F32_16X16X4_F32):
//   out[m,n] = Aagg[m,:] @ WlT[n,:] + Ax[m,:] @ WrT[n,:] + bias[n]
// (weights pre-transposed; agg pre-fixed to plain floats).
// One wave per 16x16 output tile; per k-step: 4 aligned b64 loads + 2 WMMAs
// into independent accumulators (cA/cX) summed in the epilogue.
// GUARD=false: interior tile (K%4==0, 16 valid cols) -> zero guard VALU.
// GUARD=true : branchless clamped loads + selects (EXEC stays all-1s).
// ---------------------------------------------------------------------------
template <bool GUARD>
__global__ __launch_bounds__(256) void sage_gemm_wmma(
    const float* __restrict__ Aagg, const float* __restrict__ Ax,
    const float* __restrict__ WlT, const float* __restrict__ WrT,
    const float* __restrict__ bias, float* __restrict__ out, int M, int K,
    int Nout, int tilesNLaunch, int tnBase, int do_logsoftmax) {
  const int lane = threadIdx.x & 31;
  long wid = (long)blockIdx.x * (blockDim.x >> 5) + (threadIdx.x >> 5);
  const int tilesM = (M + 15) >> 4;
  if (wid >= (long)tilesM * tilesNLaunch) return;  // wave-uniform exit
  const int tm = (int)(wid / tilesNLaunch);
  const int tn = tnBase + (int)(wid % tilesNLaunch);

  const int mrow = tm * 16 + (lane & 15);  // A: lane&15 selects M
  const int ncol = tn * 16 + (lane & 15);  // B/C/D: lane&15 selects N
  const int kadd = (lane >> 4) * 2;        // lanes 16..31 hold K = k0+2, k0+3
  const bool nok = GUARD ? (ncol < Nout) : true;

  const int mrowC = imin(mrow, M - 1);
  const int ncolC = GUARD ? imin(ncol, Nout - 1) : ncol;
  const float* __restrict__ rowA = Aagg + (long)mrowC * K;
  const float* __restrict__ rowX = Ax + (long)mrowC * K;
  const float* __restrict__ rowL = WlT + (long)ncolC * K;
  const float* __restrict__ rowR = WrT + (long)ncolC * K;

  v8f cA = {};
  v8f cX = {};
#pragma unroll 4
  for (int k0 = 0; k0 < K; k0 += 4) {
    const int ka = k0 + kadd;
    const int kp = GUARD ? imin(ka, K - 2) : ka;  // even, 8B-aligned pair base

    v2f aA = *(const v2f*)(rowA + kp);
    v2f aX = *(const v2f*)(rowX + kp);
    v2f bL = *(const v2f*)(rowL + kp);
    v2f bR = *(const v2f*)(rowR + kp);

    if (GUARD) {
      const bool k0ok = (ka < K);
      const bool k1ok = (ka + 1 < K);
      aA.x = k0ok ? aA.x : 0.0f;
      aA.y = k1ok ? aA.y : 0.0f;
      aX.x = k0ok ? aX.x : 0.0f;
      aX.y = k1ok ? aX.y : 0.0f;
      bL.x = (k0ok && nok) ? bL.x : 0.0f;
      bL.y = (k1ok && nok) ? bL.y : 0.0f;
      bR.x = (k0ok && nok) ? bR.x : 0.0f;
      bR.y = (k1ok && nok) ? bR.y : 0.0f;
    }

    // Independent accumulator chains -> WMMA streams overlap
    cA = __builtin_amdgcn_wmma_f32_16x16x4_f32(false, aA, false, bL, (short)0,
                                               cA, false, false);
    cX = __builtin_amdgcn_wmma_f32_16x16x4_f32(false, aX, false, bR, (short)0,
                                               cX, false, false);
  }

  const float bload = bias[ncolC];  // unconditional (clamped index)
  const float bn = nok ? bload : 0.0f;
  const int half = lane >> 4;  // C/D: VGPR v holds M = v (lanes 0-15) / v+8
#pragma unroll
  for (int v = 0; v < 8; ++v) {
    float val = cA[v] + cX[v] + bn;
    if (do_logsoftmax) {
      // Row (Nout <= 16) lives across the 16 lanes of this half; xor swizzles
      // with masks < 16 stay within each 16-lane half (wave32).
      float vm = nok ? val : -INFINITY;
      vm = fmaxf(vm, swz<SWZ_X1>(vm));
      vm = fmaxf(vm, swz<SWZ_X2>(vm));
      vm = fmaxf(vm, swz<SWZ_X4>(vm));
      vm = fmaxf(vm, swz<SWZ_X8>(vm));
      float ex = nok ? expf(val - vm) : 0.0f;
      ex += swz<SWZ_X1>(ex);
      ex += swz<SWZ_X2>(ex);
      ex += swz<SWZ_X4>(ex);
      ex += swz<SWZ_X8>(ex);
      val = val - vm - logf(ex);
    }
    const int m = tm * 16 + v + half * 8;
    if (nok && m < M) out[(long)m * Nout + ncol] = val;
  }
}

// ---------------------------------------------------------------------------
// Orchestration
// ---------------------------------------------------------------------------
extern "C" void kernel_launch(void* const* d_in, const int* in_sizes, int n_in,
                              void* d_out, int out_size, void* d_ws, size_t ws_size,
                              hipStream_t stream) {
  const float* x   = (const float*)d_in[0];
  const int*   ei  = (const int*)d_in[1];
  const float* Wl1 = (const float*)d_in[2];
  const float* b1  = (const float*)d_in[3];
  const float* Wr1 = (const float*)d_in[4];
  const float* Wl2 = (const float*)d_in[5];
  const float* b2  = (const float*)d_in[6];
  const float* Wr2 = (const float*)d_in[7];
  const float* Wl3 = (const float*)d_in[8];
  const float* b3  = (const float*)d_in[9];
  const float* Wr3 = (const float*)d_in[10];

  // Shapes derived from setup_inputs() sizes.
  const int H1   = in_sizes[3];        // 200
  const int H2   = in_sizes[6];        // 50
  const int C    = in_sizes[9];        // 10
  const int F_IN = in_sizes[2] / H1;   // 128
  const int N    = in_sizes[0] / F_IN; // 100000
  const int E    = in_sizes[1] / 2;    // 1600000

  const int* src = ei;
  const int* dst = ei + E;

  // Workspace (f32 elems): agg[N*H1] | h1[N*H1] | h2[N*H2] | WlT | WrT
  const long maxW = (long)F_IN * H1;  // largest weight matrix (25600)
  unsigned int* agg = (unsigned int*)d_ws;
  float* aggf = (float*)d_ws;  // same buffer after fixup (plain floats)
  float* h1  = (float*)d_ws + (long)N * H1;
  float* h2  = h1 + (long)N * H1;
  float* wlT = h2 + (long)N * H2;
  float* wrT = wlT + maxW;
  float* outp = (float*)d_out;

  auto launch_fill = [&](long n) {
    long b = (n + 255) / 256;
    int blocks = (b > (1 << 20)) ? (1 << 20) : (int)b;
    fill_kernel<<<blocks, 256, 0, stream>>>(agg, KEY_NEG_INF, n);
  };
  auto launch_scatter = [&](const float* feat, int F) {
    int blocks = (E + 7) / 8;  // 8 waves per 256-thread block, wave per edge
    scatter_max_kernel<<<blocks, 256, 0, stream>>>(feat, src, dst, agg, E, F);
  };
  auto launch_fixup = [&](long n) {
    long pairs = n >> 1;  // n is even
    long b = (pairs + 255) / 256;
    int blocks = (b > (1 << 20)) ? (1 << 20) : (int)b;
    fixup_kernel<<<blocks, 256, 0, stream>>>(agg, pairs);
  };
  auto launch_gemm = [&](const float* Ax, const float* Wl, const float* Wr,
                         const float* bias, float* out, int K, int Nout,
                         int do_ls) {
    // Pre-transpose weights into workspace (tiny).
    int wElems = K * Nout;
    int tb = (wElems + 255) / 256;
    transpose_kernel<<<tb, 256, 0, stream>>>(Wl, wlT, K, Nout);
    transpose_kernel<<<tb, 256, 0, stream>>>(Wr, wrT, K, Nout);

    const int tilesM = (N + 15) / 16;
    const int tilesN = (Nout + 15) / 16;
    const int fullN = (K % 4 == 0) ? (Nout / 16) : 0;
    const int remN = tilesN - fullN;
    if (fullN > 0) {
      long waves = (long)tilesM * fullN;
      int blocks = (int)((waves + 7) / 8);
      sage_gemm_wmma<false><<<blocks, 256, 0, stream>>>(
          aggf, Ax, wlT, wrT, bias, out, N, K, Nout, fullN, 0, do_ls);
    }
    if (remN > 0) {
      long waves = (long)tilesM * remN;
      int blocks = (int)((waves + 7) / 8);
      sage_gemm_wmma<true><<<blocks, 256, 0, stream>>>(
          aggf, Ax, wlT, wrT, bias, out, N, K, Nout, remN, fullN, do_ls);
    }
  };

  // Layer 1: agg = segmax(x); h1 = fix(agg)@Wl1 + x@Wr1 + b1
  launch_fill((long)N * F_IN);
  launch_scatter(x, F_IN);
  launch_fixup((long)N * F_IN);
  launch_gemm(x, Wl1, Wr1, b1, h1, F_IN, H1, 0);

  // Layer 2: agg = segmax(h1); h2 = fix(agg)@Wl2 + h1@Wr2 + b2
  launch_fill((long)N * H1);
  launch_scatter(h1, H1);
  launch_fixup((long)N * H1);
  launch_gemm(h1, Wl2, Wr2, b2, h2, H1, H2, 0);

  // Layer 3: agg = segmax(h2); out = log_softmax(fix(agg)@Wl3 + h2@Wr3 + b3)
  launch_fill((long)N * H2);
  launch_scatter(h2, H2);
  launch_fixup((long)N * H2);
  launch_gemm(h2, Wl3, Wr3, b3, outp, H2, C, 1);
}